// PointNet2_7825430413399
// MI455X (gfx1250) — compile-verified
//
#include <hip/hip_runtime.h>

typedef __attribute__((ext_vector_type(2))) float v2f;
typedef __attribute__((ext_vector_type(8))) float v8f;

#define BN_EPS 1e-5f

// Activation buffers use a channel-pair interleaved layout ("XP"):
//   element (c, p) of a [C][P] matrix lives at float index ((c>>1)*P + p)*2 + (c&1)
// i.e. a [C/2][P][2] array. This makes each lane's (k, k+1) WMMA source pair a
// single aligned b64 load, and epilogue row-pairs a single b64 store.

// ---------------------------------------------------------------------------
// Fold conv-bias + batchnorm into per-channel scale/shift.
// ---------------------------------------------------------------------------
__global__ void affine_kernel(const float* __restrict__ bias,
                              const float* __restrict__ gamma,
                              const float* __restrict__ beta,
                              const float* __restrict__ mean,
                              const float* __restrict__ var,
                              float* __restrict__ scale,
                              float* __restrict__ shift, int C) {
  int c = blockIdx.x * blockDim.x + threadIdx.x;
  if (c >= C) return;
  float rs = rsqrtf(var[c] + BN_EPS);
  float sc = gamma[c] * rs;
  scale[c] = sc;
  shift[c] = (bias[c] - mean[c]) * sc + beta[c];
}

// ---------------------------------------------------------------------------
// Zero-pad weights along Cin: Wp[Cout][Cinp] <- W[Cout][Cin]
// ---------------------------------------------------------------------------
__global__ void pad_w_kernel(const float* __restrict__ W, float* __restrict__ Wp,
                             int Cin, int Cinp, int Cout) {
  int i = blockIdx.x * blockDim.x + threadIdx.x;
  if (i >= Cout * Cinp) return;
  int o = i / Cinp, c = i % Cinp;
  Wp[i] = (c < Cin) ? W[(size_t)o * Cin + c] : 0.f;
}

// ---------------------------------------------------------------------------
// Farthest point sampling: one block per batch, dist in registers,
// block-wide argmax reduction in LDS. Writes sampled coords out[b][npoint][3].
// ---------------------------------------------------------------------------
__global__ void fps_kernel(const float* __restrict__ src, int bstride, int pstride,
                           int dstride, int N, int npoint,
                           float* __restrict__ out) {
  __shared__ float svals[1024];
  __shared__ int   sidx[1024];
  __shared__ float cent[3];
  const int b = blockIdx.x;
  const int tid = threadIdx.x;
  const int bd = blockDim.x;
  const float* base = src + (size_t)b * bstride;

  float dist[16];
#pragma unroll
  for (int i = 0; i < 16; ++i) dist[i] = 1e10f;

  int cur = 0;  // reference scan starts with far=0
  for (int s = 0; s < npoint; ++s) {
    if (tid == 0) {
      float cx = base[(size_t)cur * pstride + 0 * dstride];
      float cy = base[(size_t)cur * pstride + 1 * dstride];
      float cz = base[(size_t)cur * pstride + 2 * dstride];
      cent[0] = cx; cent[1] = cy; cent[2] = cz;
      out[((size_t)b * npoint + s) * 3 + 0] = cx;
      out[((size_t)b * npoint + s) * 3 + 1] = cy;
      out[((size_t)b * npoint + s) * 3 + 2] = cz;
    }
    __syncthreads();
    float cx = cent[0], cy = cent[1], cz = cent[2];

    float bestv = -1.f; int besti = 0;
    int i = 0;
    for (int n = tid; n < N; n += bd, ++i) {
      float dx = base[(size_t)n * pstride + 0 * dstride] - cx;
      float dy = base[(size_t)n * pstride + 1 * dstride] - cy;
      float dz = base[(size_t)n * pstride + 2 * dstride] - cz;
      float d = dx * dx + dy * dy + dz * dz;
      float dd = fminf(dist[i], d);
      dist[i] = dd;
      if (dd > bestv) { bestv = dd; besti = n; }
    }
    svals[tid] = bestv; sidx[tid] = besti;
    __syncthreads();
    for (int off = bd >> 1; off > 0; off >>= 1) {
      if (tid < off) {
        float v2 = svals[tid + off]; int i2 = sidx[tid + off];
        if (v2 > svals[tid] || (v2 == svals[tid] && i2 < sidx[tid])) {
          svals[tid] = v2; sidx[tid] = i2;
        }
      }
      __syncthreads();
    }
    cur = sidx[0];
    __syncthreads();
  }
}

// ---------------------------------------------------------------------------
// SA1 ball-query + gather + center-subtract. One wave32 per centroid.
// In-index-order selection via __ballot; pads with first in-radius index.
// Writes XP[4ch][B*S*K] pair-interleaved (channel 3 zero).
// ---------------------------------------------------------------------------
__global__ void group_sa1_kernel(const float* __restrict__ points,
                                 const float* __restrict__ new_xyz,
                                 float* __restrict__ X,
                                 int B_, int N, int S, int K, float r2) {
  __shared__ int idxbuf[8][64];
  const int wslot = threadIdx.x >> 5;
  const int lane = threadIdx.x & 31;
  const int wid = blockIdx.x * (blockDim.x >> 5) + wslot;
  if (wid >= B_ * S) return;
  const int b = wid / S, s = wid % S;
  const float cx = new_xyz[((size_t)b * S + s) * 3 + 0];
  const float cy = new_xyz[((size_t)b * S + s) * 3 + 1];
  const float cz = new_xyz[((size_t)b * S + s) * 3 + 2];
  const float* px = points + (size_t)b * 3 * N;

  int cnt = 0, firsti = -1;
  for (int base = 0; base < N && cnt < K; base += 32) {
    int n = base + lane;
    bool inball = false;
    if (n < N) {
      float dx = px[n] - cx, dy = px[N + n] - cy, dz = px[2 * N + n] - cz;
      inball = (dx * dx + dy * dy + dz * dz) <= r2;
    }
    unsigned long long m = __ballot(inball);
    while (m && cnt < K) {
      int bit = __ffsll((unsigned long long)m) - 1;
      int idx = base + bit;
      if (firsti < 0) firsti = idx;
      if (lane == 0) idxbuf[wslot][cnt] = idx;
      cnt++;
      m &= (m - 1);
    }
  }
  if (firsti < 0) firsti = N - 1;  // jax OOB gather clamps
  for (int t = cnt + lane; t < K; t += 32) idxbuf[wslot][t] = firsti;

  v2f* xp = (v2f*)X;
  const size_t Pt = (size_t)B_ * S * K;
  for (int t = lane; t < K; t += 32) {
    int idx = idxbuf[wslot][t];
    if (idx > N - 1) idx = N - 1;
    size_t pos = ((size_t)(b * S + s)) * K + t;
    v2f p01; p01.x = px[idx] - cx;        p01.y = px[N + idx] - cy;
    v2f p23; p23.x = px[2 * N + idx] - cz; p23.y = 0.f;  // pad channel
    xp[0 * Pt + pos] = p01;
    xp[1 * Pt + pos] = p23;
  }
}

// ---------------------------------------------------------------------------
// SA2 ball-query + gather (Dxyz ++ features). xyz1: [B][S1][3], feats1:
// [B][C1][S1] (plain). Writes XP[Cpad][B*S2*K] pair-interleaved.
// Channels: 0..2 = centered xyz, 3..3+C1-1 = features, rest zero. C1 even.
// ---------------------------------------------------------------------------
__global__ void group_sa2_kernel(const float* __restrict__ xyz1,
                                 const float* __restrict__ new_xyz2,
                                 const float* __restrict__ feats1,
                                 float* __restrict__ X,
                                 int B_, int S1, int S2, int K, int C1, int Cpad,
                                 float r2) {
  __shared__ int idxbuf[8][64];
  const int wslot = threadIdx.x >> 5;
  const int lane = threadIdx.x & 31;
  const int wid = blockIdx.x * (blockDim.x >> 5) + wslot;
  if (wid >= B_ * S2) return;
  const int b = wid / S2, s = wid % S2;
  const float cx = new_xyz2[((size_t)b * S2 + s) * 3 + 0];
  const float cy = new_xyz2[((size_t)b * S2 + s) * 3 + 1];
  const float cz = new_xyz2[((size_t)b * S2 + s) * 3 + 2];
  const float* px = xyz1 + (size_t)b * S1 * 3;

  int cnt = 0, firsti = -1;
  for (int base = 0; base < S1 && cnt < K; base += 32) {
    int n = base + lane;
    bool inball = false;
    if (n < S1) {
      float dx = px[n * 3 + 0] - cx, dy = px[n * 3 + 1] - cy, dz = px[n * 3 + 2] - cz;
      inball = (dx * dx + dy * dy + dz * dz) <= r2;
    }
    unsigned long long m = __ballot(inball);
    while (m && cnt < K) {
      int bit = __ffsll((unsigned long long)m) - 1;
      int idx = base + bit;
      if (firsti < 0) firsti = idx;
      if (lane == 0) idxbuf[wslot][cnt] = idx;
      cnt++;
      m &= (m - 1);
    }
  }
  if (firsti < 0) firsti = S1 - 1;
  for (int t = cnt + lane; t < K; t += 32) idxbuf[wslot][t] = firsti;

  v2f* xp = (v2f*)X;
  const size_t Pt = (size_t)B_ * S2 * K;
  const int npair = Cpad >> 1;
  for (int t = lane; t < K; t += 32) {
    int idx = idxbuf[wslot][t];
    if (idx > S1 - 1) idx = S1 - 1;
    size_t pos = ((size_t)(b * S2 + s)) * K + t;
    const float* fb = feats1 + (size_t)b * C1 * S1 + idx;
    v2f p01; p01.x = px[idx * 3 + 0] - cx; p01.y = px[idx * 3 + 1] - cy;
    v2f p23; p23.x = px[idx * 3 + 2] - cz; p23.y = fb[0];
    xp[0 * Pt + pos] = p01;
    xp[1 * Pt + pos] = p23;
    for (int p = 2; p < npair; ++p) {
      int c0 = 2 * p - 3;  // feature channel of first slot
      v2f q;
      q.x = fb[(size_t)c0 * S1];
      q.y = (c0 + 1 < C1) ? fb[(size_t)(c0 + 1) * S1] : 0.f;
      xp[(size_t)p * Pt + pos] = q;
    }
  }
}

// ---------------------------------------------------------------------------
// SA3 group-all: XP[Cpad][B*S] pair-interleaved; c<3 raw xyz2 (no centering),
// c<3+C2 feats2, rest zero.
// ---------------------------------------------------------------------------
__global__ void build_sa3_kernel(const float* __restrict__ xyz2,
                                 const float* __restrict__ feats2,
                                 float* __restrict__ X, int B_, int S, int C2,
                                 int Cpad) {
  int i = blockIdx.x * blockDim.x + threadIdx.x;
  int PT = B_ * S;
  int total = Cpad * PT;
  if (i >= total) return;
  int c = i / PT;
  int p = i % PT;
  int b = p / S, s = p % S;
  float v = 0.f;
  if (c < 3) v = xyz2[((size_t)b * S + s) * 3 + c];
  else if (c < 3 + C2) v = feats2[((size_t)b * C2 + (c - 3)) * S + s];
  X[((size_t)(c >> 1) * PT + p) * 2 + (c & 1)] = v;
}

// ---------------------------------------------------------------------------
// Fused pointwise-conv (GEMM) + BN + ReLU via V_WMMA_F32_16X16X4_F32.
// X and Y in pair-interleaved layout, W plain [Cout][Cin], Cin % 4 == 0.
// One wave per 16(M)x64(N) tile: per k-step 1 A b64 + 4 B b64 + 4 WMMAs.
// 32-bit WMMA src layout (wave32): VGPR0 = K=k0 (lanes 0-15) / k0+2 (16-31),
// VGPR1 = k0+1 / k0+3. C/D: VGPR j -> row mbase+j / mbase+j+8.
// ---------------------------------------------------------------------------
__global__ void wmma_gemm_kernel(const float* __restrict__ X,
                                 const float* __restrict__ W,
                                 const float* __restrict__ scale,
                                 const float* __restrict__ shift,
                                 float* __restrict__ Y,
                                 int Cin, int Cout, int P) {
  const int lane = threadIdx.x & 31;
  const int wslot = threadIdx.x >> 5;
  const int wid = blockIdx.x * (blockDim.x >> 5) + wslot;
  const int ntileN = P >> 6;            // 64 positions per wave
  const int ntiles = (Cout >> 4) * ntileN;
  if (wid >= ntiles) return;            // wave-uniform: EXEC all-1s for WMMA

  const int tm = wid / ntileN, tn = wid % ntileN;
  const int mbase = tm << 4, nbase = tn << 6;
  const int half = lane >> 4;           // 0: lanes 0-15, 1: lanes 16-31
  const int l16 = lane & 15;
  const int row = mbase + l16;          // A-matrix row (M)
  const int col = nbase + l16;          // first B/D column (N)

  v8f acc0 = {0.f, 0.f, 0.f, 0.f, 0.f, 0.f, 0.f, 0.f};
  v8f acc1 = acc0, acc2 = acc0, acc3 = acc0;

  const v2f* wp = (const v2f*)(W + (size_t)row * Cin);  // pair index 2*kb+half
  const v2f* xp = (const v2f*)X;                        // [Cin/2][P] pairs
  const int nkb = Cin >> 2;
#pragma unroll 2
  for (int kb = 0; kb < nkb; ++kb) {
    const size_t kp = (size_t)(2 * kb + half);
    v2f a  = wp[kp];
    v2f b0 = xp[kp * P + col];
    v2f b1 = xp[kp * P + col + 16];
    v2f b2 = xp[kp * P + col + 32];
    v2f b3 = xp[kp * P + col + 48];
    acc0 = __builtin_amdgcn_wmma_f32_16x16x4_f32(false, a, false, b0, (short)0, acc0, false, false);
    acc1 = __builtin_amdgcn_wmma_f32_16x16x4_f32(false, a, false, b1, (short)0, acc1, false, false);
    acc2 = __builtin_amdgcn_wmma_f32_16x16x4_f32(false, a, false, b2, (short)0, acc2, false, false);
    acc3 = __builtin_amdgcn_wmma_f32_16x16x4_f32(false, a, false, b3, (short)0, acc3, false, false);
  }

  v2f* yp = (v2f*)Y;
#pragma unroll
  for (int jp = 0; jp < 4; ++jp) {
    const int r0 = mbase + (half << 3) + 2 * jp;      // row pair (r0, r0+1)
    const size_t pr = (size_t)(r0 >> 1) * P;
    float sc0 = scale[r0], sh0 = shift[r0];
    float sc1 = scale[r0 + 1], sh1 = shift[r0 + 1];
    v8f* accs[4] = {&acc0, &acc1, &acc2, &acc3};
#pragma unroll
    for (int t = 0; t < 4; ++t) {
      v8f& A = *accs[t];
      v2f o;
      o.x = fmaxf(A[2 * jp] * sc0 + sh0, 0.f);
      o.y = fmaxf(A[2 * jp + 1] * sc1 + sh1, 0.f);
      yp[pr + col + 16 * t] = o;
    }
  }
}

// ---------------------------------------------------------------------------
// Max over K. Y in pair-interleaved layout over P = B*S*K (K fastest);
// out plain [B][C][S].
// ---------------------------------------------------------------------------
__global__ void maxpool_kernel(const float* __restrict__ Y, float* __restrict__ out,
                               int B_, int C, int S, int K) {
  int i = blockIdx.x * blockDim.x + threadIdx.x;
  if (i >= B_ * C * S) return;
  int s = i % S;
  int c = (i / S) % C;
  int b = i / (S * C);
  const size_t P = (size_t)B_ * S * K;
  const float* p = Y + ((size_t)(c >> 1) * P + (size_t)(b * S + s) * K) * 2 + (c & 1);
  float m = p[0];
  for (int k = 1; k < K; ++k) m = fmaxf(m, p[2 * k]);
  out[((size_t)b * C + c) * S + s] = m;
}

// ---------------------------------------------------------------------------
extern "C" void kernel_launch(void* const* d_in, const int* in_sizes, int n_in,
                              void* d_out, int out_size, void* d_ws, size_t ws_size,
                              hipStream_t stream) {
  (void)in_sizes; (void)n_in; (void)out_size; (void)ws_size;
  const float* points = (const float*)d_in[0];
  auto Wp  = [&](int sa, int m) { return (const float*)d_in[1  + sa * 3 + m]; };
  auto Bip = [&](int sa, int m) { return (const float*)d_in[10 + sa * 3 + m]; };
  auto Gp  = [&](int sa, int m) { return (const float*)d_in[19 + sa * 3 + m]; };
  auto Bep = [&](int sa, int m) { return (const float*)d_in[28 + sa * 3 + m]; };
  auto Mp  = [&](int sa, int m) { return (const float*)d_in[37 + sa * 3 + m]; };
  auto Vp  = [&](int sa, int m) { return (const float*)d_in[46 + sa * 3 + m]; };

  float* ws = (float*)d_ws;
  const size_t BUF = 8388608;  // 128*65536 floats (also = 256*32768)
  float* bufA   = ws;
  float* bufB   = bufA + BUF;
  float* nx1    = bufB + BUF;            // [16][128][3]
  float* nx2    = nx1 + 16 * 128 * 3;    // [16][32][3]
  float* f1     = nx2 + 16 * 32 * 3;     // [16][128][128]
  float* f2     = f1 + 16 * 128 * 128;   // [16][256][32]
  float* scaleA = f2 + 16 * 256 * 32;    // 9*1024
  float* shiftA = scaleA + 9 * 1024;
  float* wpad   = shiftA + 9 * 1024;     // max 256*260 = 66560 floats

  const int Couts[9] = {64, 64, 128, 128, 128, 256, 256, 512, 1024};
  for (int L = 0; L < 9; ++L) {
    int sa = L / 3, m = L % 3, C = Couts[L];
    affine_kernel<<<(C + 255) / 256, 256, 0, stream>>>(
        Bip(sa, m), Gp(sa, m), Bep(sa, m), Mp(sa, m), Vp(sa, m),
        scaleA + L * 1024, shiftA + L * 1024, C);
  }

  auto gemm = [&](const float* X, const float* Wt, int L, float* Yo,
                  int Cin, int Cout, int P) {
    int tiles = (Cout / 16) * (P / 64);
    wmma_gemm_kernel<<<(tiles + 7) / 8, 256, 0, stream>>>(
        X, Wt, scaleA + L * 1024, shiftA + L * 1024, Yo, Cin, Cout, P);
  };
  auto padw = [&](const float* Wt, int Cin, int Cinp, int Cout) {
    pad_w_kernel<<<(Cout * Cinp + 255) / 256, 256, 0, stream>>>(
        Wt, wpad, Cin, Cinp, Cout);
  };

  // ---- SA1 ----
  fps_kernel<<<16, 1024, 0, stream>>>(points, 3 * 16384, 1, 16384, 16384, 128, nx1);
  group_sa1_kernel<<<(16 * 128 + 7) / 8, 256, 0, stream>>>(
      points, nx1, bufA, 16, 16384, 128, 32, 0.2f * 0.2f);
  const int P1 = 16 * 128 * 32;  // 65536
  padw(Wp(0, 0), 3, 4, 64);
  gemm(bufA, wpad, 0, bufB, 4, 64, P1);
  gemm(bufB, Wp(0, 1), 1, bufA, 64, 64, P1);
  gemm(bufA, Wp(0, 2), 2, bufB, 64, 128, P1);
  maxpool_kernel<<<(16 * 128 * 128 + 255) / 256, 256, 0, stream>>>(
      bufB, f1, 16, 128, 128, 32);

  // ---- SA2 ----
  fps_kernel<<<16, 128, 0, stream>>>(nx1, 128 * 3, 3, 1, 128, 32, nx2);
  group_sa2_kernel<<<(16 * 32 + 7) / 8, 256, 0, stream>>>(
      nx1, nx2, f1, bufA, 16, 128, 32, 64, 128, 132, 0.4f * 0.4f);
  const int P2 = 16 * 32 * 64;  // 32768
  padw(Wp(1, 0), 131, 132, 128);
  gemm(bufA, wpad, 3, bufB, 132, 128, P2);
  gemm(bufB, Wp(1, 1), 4, bufA, 128, 128, P2);
  gemm(bufA, Wp(1, 2), 5, bufB, 128, 256, P2);
  maxpool_kernel<<<(16 * 256 * 32 + 255) / 256, 256, 0, stream>>>(
      bufB, f2, 16, 256, 32, 64);

  // ---- SA3 (group all) ----
  build_sa3_kernel<<<(260 * 512 + 255) / 256, 256, 0, stream>>>(
      nx2, f2, bufA, 16, 32, 256, 260);
  const int P3 = 16 * 32;  // 512
  padw(Wp(2, 0), 259, 260, 256);
  gemm(bufA, wpad, 6, bufB, 260, 256, P3);
  gemm(bufB, Wp(2, 1), 7, bufA, 256, 512, P3);
  gemm(bufA, Wp(2, 2), 8, bufB, 512, 1024, P3);
  maxpool_kernel<<<(16 * 1024 + 255) / 256, 256, 0, stream>>>(
      bufB, (float*)d_out, 16, 1024, 1, 32);
}